// RotaryPositionalEmbedding_26517128086003
// MI455X (gfx1250) — compile-verified
//
#include <hip/hip_runtime.h>
#include <stdint.h>

// Problem constants (from the reference)
#define D_K      128
#define HALF     64        // D_K/2 rotation pairs
#define SEQ_LEN  4096
#define BH       128       // BATCH*HEADS = 4*32
#define S_TILE   32        // positions per block
#define TPB      256       // 8 wave32 per block

typedef float v4f __attribute__((ext_vector_type(4)));

// ---------------------------------------------------------------------------
// Kernel 1: gather the cos/sin diagonals of R into a compact, interleaved
// table cs[s][d]  (cs[s][2k]=cos, cs[s][2k+1]=sin).  4096*128 fp32 = 2 MiB,
// L2-resident, reused 128x (over batch*heads) by kernel 2.
// ---------------------------------------------------------------------------
__global__ __launch_bounds__(HALF) void rope_gather_cs(
    const float* __restrict__ R,
    const int*   __restrict__ pos,
    float*       __restrict__ cs) {
  const int s = blockIdx.x;          // sequence index
  const int k = threadIdx.x;         // rotation pair 0..63
  const long base = (long)pos[s] * (D_K * D_K);
  const float c  = R[base + (2 * k)     * D_K + 2 * k];   // R[p][2k][2k]
  const float sn = R[base + (2 * k + 1) * D_K + 2 * k];   // R[p][2k+1][2k]
  cs[s * D_K + 2 * k]     = c;
  cs[s * D_K + 2 * k + 1] = sn;
}

// ---------------------------------------------------------------------------
// Kernel 2: out = rotate(x).  Each block handles one (b,h) and 32 contiguous
// positions -> a contiguous 16 KiB chunk of x/out.  The matching 16 KiB cs
// tile is staged into LDS with gfx1250 async global->LDS b128 copies
// (tracked by ASYNCcnt), then the x stream is processed with non-temporal
// b128 loads/stores so it does not evict the cs table from L2.
// ---------------------------------------------------------------------------
__global__ __launch_bounds__(TPB) void rope_apply(
    const float* __restrict__ x,
    const float* __restrict__ cs,
    float*       __restrict__ out) {
  __shared__ float smem[S_TILE * D_K];   // 16 KiB cs tile

  const int tid    = threadIdx.x;
  const int bh     = blockIdx.x;
  const int s_base = blockIdx.y * S_TILE;

  // ---- async stage: 1024 float4 (16 KiB), 4 b128 copies per thread --------
  const float* cs_blk = cs + (size_t)s_base * D_K;   // uniform -> SGPR base
  // LDS byte address of smem[0]: low 32 bits of the generic LDS address are
  // the LDS offset on gfx1250 (aperture lives in the high 32 bits).
  const uint32_t smem_base = (uint32_t)(uintptr_t)(void*)&smem[0];

#pragma unroll
  for (int it = 0; it < 4; ++it) {
    const uint32_t f    = (uint32_t)(it * TPB + tid);  // float4 index 0..1023
    const uint32_t goff = f * 16u;                     // per-lane byte offset
    const uint32_t ldsb = smem_base + f * 16u;         // per-lane LDS dest
    asm volatile("global_load_async_to_lds_b128 %0, %1, %2"
                 :: "v"(ldsb), "v"(goff), "s"(cs_blk)
                 : "memory");
  }
  // Each wave drains its own ASYNCcnt, then the workgroup barrier makes the
  // LDS tile visible to all 8 waves.
  asm volatile("s_wait_asynccnt 0" ::: "memory");
  __syncthreads();

  // ---- streaming rotate: 4 float4 per thread ------------------------------
  const size_t row0 = ((size_t)bh * SEQ_LEN + (size_t)s_base) * D_K; // floats
  const v4f* __restrict__ xin  = (const v4f*)(x + row0);
  v4f*       __restrict__ xout = (v4f*)(out + row0);
  const v4f* __restrict__ csl  = (const v4f*)smem;

#pragma unroll
  for (int it = 0; it < 4; ++it) {
    const int f = it * TPB + tid;                 // float4 idx in tile
    v4f xv = __builtin_nontemporal_load(&xin[f]); // {x0,x1,x2,x3}
    v4f t  = csl[f];                              // {c0,s0,c1,s1}
    v4f o;
    o.x = t.x * xv.x - t.y * xv.y;                // c0*x0 - s0*x1
    o.y = t.y * xv.x + t.x * xv.y;                // s0*x0 + c0*x1
    o.z = t.z * xv.z - t.w * xv.w;                // c1*x2 - s1*x3
    o.w = t.w * xv.z + t.z * xv.w;                // s1*x2 + c1*x3
    __builtin_nontemporal_store(o, &xout[f]);
  }
}

// ---------------------------------------------------------------------------
extern "C" void kernel_launch(void* const* d_in, const int* in_sizes, int n_in,
                              void* d_out, int out_size, void* d_ws, size_t ws_size,
                              hipStream_t stream) {
  const float* x   = (const float*)d_in[0];  // [4,32,4096,128] fp32
  const int*   pos = (const int*)  d_in[1];  // [4096] int32
  const float* R   = (const float*)d_in[2];  // [4096,128,128] fp32
  float*       out = (float*)d_out;
  float*       cs  = (float*)d_ws;           // SEQ_LEN*D_K fp32 = 2 MiB scratch

  rope_gather_cs<<<dim3(SEQ_LEN), dim3(HALF), 0, stream>>>(R, pos, cs);
  rope_apply<<<dim3(BH, SEQ_LEN / S_TILE), dim3(TPB), 0, stream>>>(x, cs, out);
}